// LCN_4698694222614
// MI455X (gfx1250) — compile-verified
//
#include <hip/hip_runtime.h>

typedef __attribute__((ext_vector_type(2))) float v2f;
typedef __attribute__((ext_vector_type(8))) float v8f;

#define B_SZ 512
#define BQ   128   // B_SZ / 4 (float4 quads per activation row)

// ---------------------------------------------------------------------------
// Transpose x (rows=512, cols=16384) -> xT (cols, rows). Classic LDS tile.
// ---------------------------------------------------------------------------
__global__ __launch_bounds__(256) void transpose_k(const float* __restrict__ in,
                                                   float* __restrict__ out,
                                                   int rows, int cols) {
    __shared__ float t[32][33];
    const int x  = blockIdx.x * 32 + threadIdx.x;   // input col
    const int y0 = blockIdx.y * 32;                 // input row base
    #pragma unroll
    for (int i = threadIdx.y; i < 32; i += 8)
        t[i][threadIdx.x] = in[(y0 + i) * cols + x];
    __syncthreads();
    const int xo  = y0 + threadIdx.x;               // output col (= input row)
    const int yo0 = blockIdx.x * 32;                // output row base (= input col)
    #pragma unroll
    for (int i = threadIdx.y; i < 32; i += 8)
        out[(yo0 + i) * rows + xo] = t[threadIdx.x][i];
}

// ---------------------------------------------------------------------------
// One locally-connected layer, all-transposed layout:
//   xT:   (prev_dim, 512)  outT: (d_total, 512)
//   outT[d][b] = relu(bias[d] + sum_k xT[knn[d,k]][b] * w[d,k])
// Thread = (neuron d, batch quad). Wave = 32 lanes x float4 = 128 batch cols.
// The knn/w row base is wave-uniform but derived from threadIdx, so we pin it
// to an SGPR with readfirstlane -> backend emits wide s_load (SMEM/KMcnt) for
// indices+weights, keeping VMEM purely for the coalesced b128 gathers.
// Grid: (d_total/8, 4), block 256 (8 waves, one neuron per wave).
// ---------------------------------------------------------------------------
__global__ __launch_bounds__(256) void lcn_layer(const float* __restrict__ xT,
                                                 const int*   __restrict__ knn,
                                                 const float* __restrict__ w,
                                                 const float* __restrict__ bias,
                                                 float* __restrict__ outT) {
    const int lane = threadIdx.x & 31;
    const int wave = threadIdx.x >> 5;
    const int d    = __builtin_amdgcn_readfirstlane(blockIdx.x * 8 + wave);
    const int bq   = blockIdx.y * 32 + lane;        // batch-quad index [0,128)

    const float4* __restrict__ x4 = (const float4*)xT;
    const int base = __builtin_amdgcn_readfirstlane(d * 32);

    float ax = 0.f, ay = 0.f, az = 0.f, aw = 0.f;
    #pragma unroll
    for (int k = 0; k < 32; ++k) {
        const int   idx = knn[base + k];            // uniform addr -> s_load
        const float wv  = w[base + k];              // uniform addr -> s_load
        const float4 xv = x4[(size_t)idx * BQ + bq];
        ax = fmaf(wv, xv.x, ax);
        ay = fmaf(wv, xv.y, ay);
        az = fmaf(wv, xv.z, az);
        aw = fmaf(wv, xv.w, aw);
    }
    const float bv = bias[d];
    float4 r;
    r.x = fmaxf(ax + bv, 0.f);
    r.y = fmaxf(ay + bv, 0.f);
    r.z = fmaxf(az + bv, 0.f);
    r.w = fmaxf(aw + bv, 0.f);
    ((float4*)outT)[(size_t)d * BQ + bq] = r;
}

// ---------------------------------------------------------------------------
// Final FC with f32 WMMA:  out(512,16) = act3(512,2048) @ fc_w(2048,16) + fc_b
// act3 arrives transposed as xT (2048, 512).
// Each wave: one 16(batch) x 16(out) tile, 512 steps of V_WMMA_F32_16X16X4_F32.
// A layout (16x4 f32): v0 = K=0 (lanes 0-15) / K=2 (lanes 16-31); v1 = K=1/K=3.
// B layout (4x16 f32): v0 = row K=0 (lanes 0-15) / K=2 (lanes 16-31); v1 = K=1/K=3.
// D layout: VGPR r -> (M=r, N=lane) lanes 0-15, (M=r+8, N=lane-16) lanes 16-31.
// Grid: 32 blocks x 32 threads (one wave each).
// ---------------------------------------------------------------------------
__global__ __launch_bounds__(32) void fc_wmma(const float* __restrict__ xT,
                                              const float* __restrict__ fcw,
                                              const float* __restrict__ fcb,
                                              float* __restrict__ out) {
    const int lane  = threadIdx.x & 31;
    const int half  = lane >> 4;       // 0: K even pair base 0 ; 1: base 2
    const int l16   = lane & 15;
    const int mBase = blockIdx.x * 16; // batch tile

    v8f c = {};
    for (int kb = 0; kb < 2048; kb += 4) {
        const int k0 = kb + half * 2;
        v2f a, b;
        a.x = xT[(size_t)(k0    ) * B_SZ + mBase + l16];
        a.y = xT[(size_t)(k0 + 1) * B_SZ + mBase + l16];
        b.x = fcw[(k0    ) * 16 + l16];
        b.y = fcw[(k0 + 1) * 16 + l16];
        c = __builtin_amdgcn_wmma_f32_16x16x4_f32(
                /*neg_a=*/false, a, /*neg_b=*/false, b,
                /*c_mod=*/(short)0, c, /*reuse_a=*/false, /*reuse_b=*/false);
    }
    const float bn = fcb[l16];
    #pragma unroll
    for (int r = 0; r < 8; ++r) {
        const int m = mBase + half * 8 + r;
        out[m * 16 + l16] = c[r] + bn;
    }
}

// ---------------------------------------------------------------------------
// Input order: x, w0,b0,knn0, w1,b1,knn1, w2,b2,knn2, fc_w, fc_b
// ---------------------------------------------------------------------------
extern "C" void kernel_launch(void* const* d_in, const int* in_sizes, int n_in,
                              void* d_out, int out_size, void* d_ws, size_t ws_size,
                              hipStream_t stream) {
    (void)in_sizes; (void)n_in; (void)out_size; (void)ws_size;

    const float* x    = (const float*)d_in[0];
    const float* w0   = (const float*)d_in[1];
    const float* b0   = (const float*)d_in[2];
    const int*   knn0 = (const int*)  d_in[3];
    const float* w1   = (const float*)d_in[4];
    const float* b1   = (const float*)d_in[5];
    const int*   knn1 = (const int*)  d_in[6];
    const float* w2   = (const float*)d_in[7];
    const float* b2   = (const float*)d_in[8];
    const int*   knn2 = (const int*)  d_in[9];
    const float* fcw  = (const float*)d_in[10];
    const float* fcb  = (const float*)d_in[11];
    float*       out  = (float*)d_out;

    char* ws = (char*)d_ws;
    float* xT0 = (float*)(ws);                         // 16384*512*4 = 32 MB
    float* a1  = (float*)(ws + (size_t)33554432);      // 8192*512*4  = 16 MB
    float* a2  = (float*)(ws);                         // 4096*512*4  =  8 MB (xT0 dead)
    float* a3  = (float*)(ws + (size_t)8388608);       // 2048*512*4  =  4 MB

    // 1) transpose input to (16384, 512)
    transpose_k<<<dim3(16384 / 32, B_SZ / 32), dim3(32, 8), 0, stream>>>(
        x, xT0, B_SZ, 16384);

    // 2) three locally-connected layers, transposed activations throughout
    lcn_layer<<<dim3(8192 / 8, 4), 256, 0, stream>>>(xT0, knn0, w0, b0, a1);
    lcn_layer<<<dim3(4096 / 8, 4), 256, 0, stream>>>(a1,  knn1, w1, b1, a2);
    lcn_layer<<<dim3(2048 / 8, 4), 256, 0, stream>>>(a2,  knn2, w2, b2, a3);

    // 3) dense FC via f32 WMMA -> (512, 16) row-major output
    fc_wmma<<<dim3(B_SZ / 16), dim3(32), 0, stream>>>(a3, fcw, fcb, out);
}